// CDLoss_49864570306616
// MI455X (gfx1250) — compile-verified
//
#include <hip/hip_runtime.h>
#include <math.h>

// CDNA5 / gfx1250: wave32, WMMA 16x16x4 f32
typedef float v2f __attribute__((ext_vector_type(2)));
typedef float v8f __attribute__((ext_vector_type(8)));

#define TILE 16
#define CHUNK 2048            // columns of B staged in LDS per block
#define WAVES_PER_BLOCK 8
#define THREADS (WAVES_PER_BLOCK * 32)

static __device__ __forceinline__ unsigned umin32(unsigned a, unsigned b) {
    return a < b ? a : b;
}

// For each row a of A (16-row strip per wave), min over staged columns b of B of
//   d^2 = a^2 + b^2 - 2 a.b, computed ENTIRELY inside one WMMA:
//   A' = (-2ax, -2ay, -2az, 1) (spare K slot), B' = (bx, by, bz, b^2),
//   C  = broadcast of a^2 per row  =>  D = d^2 >= 0.
// Since d^2 >= 0, uint bit-pattern order == float order, so the running min is
// v_min_u32/v_min3_u32 (no float canonicalization). Cross-block combine via
// global atomicMin on the uint pattern.
__global__ __launch_bounds__(THREADS)
void cd_min_pass(const float* __restrict__ A, const float* __restrict__ B,
                 unsigned* __restrict__ outMin, int NA, int NB)
{
    __shared__ float tile[CHUNK * 4];   // per point: x, y, z, |b|^2  (32 KB)

    const int tid  = threadIdx.x;
    const int wave = tid >> 5;
    const int lane = tid & 31;
    const int half = lane >> 4;      // 0: K0,K1   1: K2,K3
    const int idx  = lane & 15;

    // ---- stage B chunk into LDS, fully uniform loop (EXEC all 1s) ----
    const int cBase = blockIdx.x * CHUNK;
    #pragma unroll
    for (int k = tid; k < CHUNK; k += THREADS) {
        const float* p = B + 3 * (cBase + k);
        float x = p[0], y = p[1], z = p[2];
        ((float4*)tile)[k] = make_float4(x, y, z, x * x + y * y + z * z);
    }
    __syncthreads();

    // ---- A fragment: (-2ax,-2ay | -2az, 1) ----
    const int r0 = (blockIdx.y * WAVES_PER_BLOCK + wave) * TILE;
    const float* pa = A + 3 * (r0 + idx);
    const float ax = pa[0], ay = pa[1], az = pa[2];
    v2f av;
    av.x = half ? (-2.0f * az) : (-2.0f * ax);
    av.y = half ? 1.0f         : (-2.0f * ay);

    // ---- C fragment: a^2 broadcast per row (VGPR v holds row v + 8*half) ----
    const float asq_own = ax * ax + ay * ay + az * az;   // row r0+idx (both halves)
    v8f casq;
    #pragma unroll
    for (int v = 0; v < 8; ++v)
        casq[v] = __shfl(asq_own, 8 * half + v, 32);

    unsigned rowmin[8];
    #pragma unroll
    for (int v = 0; v < 8; ++v) rowmin[v] = 0x7F800000u;   // +inf

    // ---- hot loop: ds_load_b64 + wmma (full d^2) + integer min ----
    const v2f* lds2 = (const v2f*)tile;
    const int laneOff = idx * 2 + half;
    #pragma unroll 4
    for (int t = 0; t < CHUNK / TILE; ++t) {
        v2f bv = lds2[t * 32 + laneOff];     // half0: (x,y)  half1: (z, b^2)
        v8f d = __builtin_amdgcn_wmma_f32_16x16x4_f32(
                /*neg_a=*/false, av, /*neg_b=*/false, bv,
                /*c_mod=*/(short)0, casq, /*reuse_a=*/false, /*reuse_b=*/false);
        #pragma unroll
        for (int v = 0; v < 8; ++v)
            rowmin[v] = umin32(rowmin[v], __float_as_uint(d[v]));   // d^2 >= 0
    }

    // ---- reduce mins across the 16 lanes (columns) of each half-wave ----
    #pragma unroll
    for (int m = 8; m >= 1; m >>= 1) {
        #pragma unroll
        for (int v = 0; v < 8; ++v)
            rowmin[v] = umin32(rowmin[v],
                               (unsigned)__shfl_xor((int)rowmin[v], m, 32));
    }

    // lanes 0 and 16 own rows r0+v (+8 for half 1): clamp, combine globally
    if (idx == 0) {
        #pragma unroll
        for (int v = 0; v < 8; ++v) {
            int row = r0 + v + 8 * half;
            float val = fmaxf(__uint_as_float(rowmin[v]), 0.0f);  // jnp.maximum(sq,0)
            atomicMin(&outMin[row], __float_as_uint(val));
        }
    }
}

__global__ void cd_init(unsigned* __restrict__ buf, int n)
{
    int i = blockIdx.x * blockDim.x + threadIdx.x;
    if (i < n) buf[i] = 0x7F800000u;   // +inf
}

// mean(sqrt(minsA)) + mean(sqrt(minsB)) -> out[0]
__global__ __launch_bounds__(256)
void cd_reduce(const unsigned* __restrict__ minsA, int nA,
               const unsigned* __restrict__ minsB, int nB,
               float* __restrict__ out)
{
    float acc = 0.0f;
    const float sA = 1.0f / (float)nA;
    const float sB = 1.0f / (float)nB;
    for (int i = threadIdx.x; i < nA; i += 256)
        acc += sqrtf(__uint_as_float(minsA[i])) * sA;
    for (int i = threadIdx.x; i < nB; i += 256)
        acc += sqrtf(__uint_as_float(minsB[i])) * sB;

    #pragma unroll
    for (int m = 16; m >= 1; m >>= 1)
        acc += __shfl_xor(acc, m, 32);

    __shared__ float wsum[8];
    int wave = threadIdx.x >> 5, lane = threadIdx.x & 31;
    if (lane == 0) wsum[wave] = acc;
    __syncthreads();
    if (threadIdx.x == 0) {
        float t = 0.0f;
        #pragma unroll
        for (int w = 0; w < 8; ++w) t += wsum[w];
        out[0] = t;
    }
}

extern "C" void kernel_launch(void* const* d_in, const int* in_sizes, int n_in,
                              void* d_out, int out_size, void* d_ws, size_t ws_size,
                              hipStream_t stream)
{
    const float* pred = (const float*)d_in[0];   // [N,3] f32
    const float* gt   = (const float*)d_in[1];   // [M,3] f32
    const int NA = in_sizes[0] / 3;              // 16384
    const int NB = in_sizes[1] / 3;              // 16384

    unsigned* minPred = (unsigned*)d_ws;         // per-pred-point min sq dist (dist2)
    unsigned* minGT   = minPred + NA;            // per-gt-point  min sq dist (dist1)

    cd_init<<<(NA + NB + 255) / 256, 256, 0, stream>>>(minPred, NA + NB);

    dim3 block(THREADS);
    // dist2: rows = pred, cols = gt
    dim3 gridP((unsigned)(NB / CHUNK), (unsigned)(NA / (TILE * WAVES_PER_BLOCK)));
    cd_min_pass<<<gridP, block, 0, stream>>>(pred, gt, minPred, NA, NB);
    // dist1: rows = gt, cols = pred
    dim3 gridG((unsigned)(NA / CHUNK), (unsigned)(NB / (TILE * WAVES_PER_BLOCK)));
    cd_min_pass<<<gridG, block, 0, stream>>>(gt, pred, minGT, NB, NA);

    cd_reduce<<<1, 256, 0, stream>>>(minPred, NA, minGT, NB, (float*)d_out);
}